// CTCDecoder_42159398978105
// MI455X (gfx1250) — compile-verified
//
#include <hip/hip_runtime.h>
#include <math.h>

typedef __attribute__((ext_vector_type(2))) float v2f;
typedef __attribute__((ext_vector_type(8))) float v8f;

constexpr int kB = 32, kT = 512, kD = 512, kV = 1024, kL = 128;
constexpr int kS = 2 * kL + 1;           // 257 extended states
constexpr float kNEG = -1e30f;

__device__ __forceinline__ float lae(float x, float y) {
  // log(exp(x)+exp(y)) safe for kNEG sentinels
  float m = fmaxf(x, y);
  float n = fminf(x, y);
  if (m <= -1e29f) return kNEG;
  return m + log1pf(__expf(n - m));
}

// -----------------------------------------------------------------------------
// Kernel 1: fused fp32-WMMA GEMM + bias + log-softmax + extended-label gather.
// grid = (B*T)/16 blocks of 256 threads (8 waves). Each block owns 16 rows
// (one b, 16 consecutive t) and the FULL V=1024 columns, so row-wise
// logsumexp needs no cross-block communication. Logits tile lives in LDS.
// Output: logp_ext[t][b][s] = log_softmax(logits)[b,t,ext[b,s]]
// -----------------------------------------------------------------------------
__global__ void fused_logits_lse_gather(
    const float* __restrict__ x, const float* __restrict__ W,
    const float* __restrict__ bias, const int* __restrict__ target,
    float* __restrict__ logp_ext)
{
  extern __shared__ float smem[];
  float* xt  = smem;               // [16*kD]  x row tile (32 KB)
  float* lg  = xt + 16 * kD;       // [16*kV]  logits tile (64 KB)
  float* red = lg + 16 * kV;       // [256]    reduction scratch
  float* lse = red + 256;          // [16]     per-row max, then logsumexp
  int*  tlab = (int*)(lse + 16);   // [kL]     target labels of this batch row

  const int tid  = threadIdx.x;
  const int wave = tid >> 5;
  const int lane = tid & 31;
  const int m     = lane & 15;        // A row / D col within tile
  const int half  = lane >> 4;
  const int khalf = half * 2;         // K sub-offset per ISA A/B layout
  const int r0 = blockIdx.x * 16;     // flat row = b*T + t
  const int bb = r0 >> 9;             // / kT
  const int t0 = r0 & (kT - 1);

  // Stage x tile (rows are contiguous in memory -> one 32KB linear copy)
  {
    const float4* xs = (const float4*)(x + (size_t)r0 * kD);
    float4* xd = (float4*)xt;
    for (int i = tid; i < 16 * kD / 4; i += 256) xd[i] = xs[i];
    for (int i = tid; i < kL; i += 256) tlab[i] = target[bb * kL + i];
  }
  __syncthreads();

  // 8 accumulator tiles per wave: wave w owns col tiles (w*8+j)*16
  v8f acc[8];
  #pragma unroll
  for (int j = 0; j < 8; ++j)
    #pragma unroll
    for (int r = 0; r < 8; ++r) acc[j][r] = 0.0f;

  for (int k0 = 0; k0 < kD; k0 += 4) {
    // A 16x4 fp32: lanes 0-15 hold K={k0,k0+1}, lanes 16-31 K={k0+2,k0+3}
    v2f a = *(const v2f*)&xt[m * kD + k0 + khalf];
    #pragma unroll
    for (int j = 0; j < 8; ++j) {
      const int n = ((wave * 8 + j) << 4) + m;
      v2f bv;
      bv.x = W[(size_t)(k0 + khalf    ) * kV + n];   // B[k][n], coalesced in n
      bv.y = W[(size_t)(k0 + khalf + 1) * kV + n];
      acc[j] = __builtin_amdgcn_wmma_f32_16x16x4_f32(
          false, a, false, bv, (short)0, acc[j], false, false);
    }
  }

  // bias add + spill logits tile to LDS (D layout: M = r + 8*half, N = m)
  #pragma unroll
  for (int j = 0; j < 8; ++j) {
    const int n = ((wave * 8 + j) << 4) + m;
    const float bs = bias[n];
    #pragma unroll
    for (int r = 0; r < 8; ++r)
      lg[(r + 8 * half) * kV + n] = acc[j][r] + bs;
  }
  __syncthreads();

  // per-row logsumexp over kV: 16 threads per row
  const int rid = tid >> 4, cid = tid & 15;
  float mx = -1e38f;
  for (int c = cid; c < kV; c += 16) mx = fmaxf(mx, lg[rid * kV + c]);
  red[rid * 16 + cid] = mx;
  __syncthreads();
  if (cid == 0) {
    float r = red[rid * 16];
    #pragma unroll
    for (int i = 1; i < 16; ++i) r = fmaxf(r, red[rid * 16 + i]);
    lse[rid] = r;
  }
  __syncthreads();
  const float rmax = lse[rid];
  float sm = 0.0f;
  for (int c = cid; c < kV; c += 16) sm += __expf(lg[rid * kV + c] - rmax);
  __syncthreads();
  red[rid * 16 + cid] = sm;
  __syncthreads();
  if (cid == 0) {
    float r = 0.0f;
    #pragma unroll
    for (int i = 0; i < 16; ++i) r += red[rid * 16 + i];
    lse[rid] = rmax + logf(r);
  }
  __syncthreads();

  // gather log-probs of extended label sequence: ext[s] = s odd ? tgt[s/2] : 0
  for (int idx = tid; idx < 16 * kS; idx += 256) {
    const int row = idx / kS;
    const int s = idx - row * kS;
    const int label = (s & 1) ? tlab[s >> 1] : 0;
    const float v = lg[row * kV + label] - lse[row];
    logp_ext[((size_t)(t0 + row) * kB + bb) * kS + s] = v;
  }
}

// -----------------------------------------------------------------------------
// Kernel 2: CTC alpha recursion. One block per batch element, 288 threads so
// all 257 states step in parallel; double-buffered alpha in LDS, one barrier
// per timestep (the serial chain of 511 steps is the true bottleneck).
// -----------------------------------------------------------------------------
__global__ void ctc_alpha_kernel(
    const float* __restrict__ logp_ext, const int* __restrict__ target,
    const int* __restrict__ in_len, const int* __restrict__ tg_len,
    float* __restrict__ out)
{
  __shared__ float abuf[2][kS + 2];   // front-padded by 2 with kNEG
  const int tid = threadIdx.x;
  const int bb  = blockIdx.x;
  const int Tin = in_len[bb];
  const int Lb  = tg_len[bb];
  const int s   = tid;

  // skip transition allowed iff ext[s]!=blank && ext[s]!=ext[s-2] && s>=2.
  // Odd s carry labels in [1,V), even s are blank.
  bool skip = false;
  if (s < kS && (s & 1) && s >= 2) {
    const int cl = target[bb * kL + (s >> 1)];
    const int pl = target[bb * kL + (s >> 1) - 1];
    skip = (cl != pl);
  }

  if (tid < kS + 2) { abuf[0][tid] = kNEG; abuf[1][tid] = kNEG; }
  __syncthreads();
  if (s == 0) abuf[0][2] = logp_ext[(size_t)bb * kS + 0];
  if (s == 1) abuf[0][3] = logp_ext[(size_t)bb * kS + 1];
  __syncthreads();

  int cur = 0;
  for (int t = 1; t < Tin; ++t) {
    float nv = kNEG;
    if (s < kS) {
      float c = lae(abuf[cur][s + 2], abuf[cur][s + 1]);
      if (skip) c = lae(c, abuf[cur][s]);
      nv = c + logp_ext[((size_t)t * kB + bb) * kS + s];
    }
    if (s < kS) abuf[cur ^ 1][s + 2] = nv;   // write other buffer: no race
    cur ^= 1;
    __syncthreads();
  }

  if (tid == 0) {
    const int s1 = 2 * Lb - 1, s2 = 2 * Lb;
    const float a1 = (s1 >= 0 && s1 < kS) ? abuf[cur][s1 + 2] : kNEG;
    const float a2 = (s2 >= 0 && s2 < kS) ? abuf[cur][s2 + 2] : kNEG;
    float nll = -lae(a1, a2);
    if (isinf(nll) || nll > 1e29f) nll = 0.0f;   // zero_infinity
    const int denom = (Lb > 0) ? Lb : 1;
    atomicAdd(out, nll / (float)denom / (float)kB);
  }
}

__global__ void zero_out_kernel(float* __restrict__ out) {
  if (threadIdx.x == 0 && blockIdx.x == 0) out[0] = 0.0f;
}

// -----------------------------------------------------------------------------
extern "C" void kernel_launch(void* const* d_in, const int* in_sizes, int n_in,
                              void* d_out, int out_size, void* d_ws, size_t ws_size,
                              hipStream_t stream) {
  (void)in_sizes; (void)n_in; (void)out_size; (void)ws_size;
  const float* x      = (const float*)d_in[0];
  const float* W      = (const float*)d_in[1];
  const float* bias   = (const float*)d_in[2];
  const int*   target = (const int*)d_in[3];
  const int*   in_len = (const int*)d_in[4];
  const int*   tg_len = (const int*)d_in[5];
  float* out = (float*)d_out;
  float* logp_ext = (float*)d_ws;   // [T][B][S] fp32 = 16.9 MB scratch

  const size_t lds_bytes =
      (size_t)(16 * kD + 16 * kV + 256 + 16) * sizeof(float) + kL * sizeof(int);
  // CDNA5 allows up to 320KB LDS per workgroup; raise the HIP cap (no-op if
  // unsupported). Deterministic per call; not a stream op, graph-capture safe.
  (void)hipFuncSetAttribute(reinterpret_cast<const void*>(fused_logits_lse_gather),
                            hipFuncAttributeMaxDynamicSharedMemorySize,
                            (int)lds_bytes);

  zero_out_kernel<<<1, 32, 0, stream>>>(out);
  fused_logits_lse_gather<<<dim3((kB * kT) / 16), dim3(256), lds_bytes, stream>>>(
      x, W, bias, target, logp_ext);
  ctc_alpha_kernel<<<dim3(kB), dim3(288), 0, stream>>>(
      logp_ext, target, in_len, tg_len, out);
}